// HungarianMatcher_Primitive_91259465105841
// MI455X (gfx1250) — compile-verified
//
#include <hip/hip_runtime.h>
#include <hip/hip_bf16.h>

// HungarianMatcher cost-matrix assembly for MI455X (gfx1250, wave32).
//
// C[bq,t] = sum_k |pred_shape[bq,k]-tgt_shape[t,k]|  -  softmax(logits)[bq, label[t]]
//
// Output 32000x3200 f32 = 409.6MB -> store-bandwidth bound (~17.6us at 23.3TB/s);
// per-element VALU work sits at roughly the same roofline, so the class term
// (an exact one-hot matmul: prob[16x92] x onehot[92x16]) is offloaded to the
// matrix pipe as 23 V_WMMA_F32_16X16X4_F32 k-steps (exact in f32: one-hot =>
// FMAs by literal 0.0), co-executing with the VALU L1-distance computation that
// pre-fills the WMMA accumulators.
//
// v3 changes (fix the v2 register blow-up seen in the histogram):
//  - k-step loop is unroll(disable)d: v2's full unroll kept ~460 VGPRs live,
//    pushing the allocator into the extended-VGPR space (s_set_vgpr_msb after
//    nearly every VALU op, 1-2 waves/SIMD). Now live state is ~4x v8f acc + temps
//    -> <128 VGPRs, no MSB churn, enough waves to cover the store stream.
//  - one-hot compares test precomputed (lab - 2*hi) / (lab - 2*hi - 1) against the
//    uniform SGPR loop counter: 2x v_cmp + 2x v_cndmask per k-step, no address math.

typedef __attribute__((ext_vector_type(2))) float v2f;
typedef __attribute__((ext_vector_type(4))) float v4f;
typedef __attribute__((ext_vector_type(8))) float v8f;

#define BSQ          32000   // 32 * 1000 flattened rows
#define TCOLS        3200
#define NCLS         92      // 23 * 4  -> 23 WMMA k-steps of K=4
#define ROWS_PER_BLK 64
#define COLS_PER_BLK 640
#define PROB_PITCH   98      // even (8B-aligned b64 A-frag loads), 98%64=34 -> conflict-free rows

__global__ __launch_bounds__(256)
void hungarian_cost_kernel(const float* __restrict__ logits,   // [BSQ, 92]
                           const float* __restrict__ predS,    // [BSQ, 4]
                           const float* __restrict__ tgtS,     // [TCOLS, 4]
                           const int*   __restrict__ tgtL,     // [TCOLS]
                           float* __restrict__ out)            // [BSQ, TCOLS]
{
    __shared__ __align__(16) float sProb[ROWS_PER_BLK][PROB_PITCH]; // negated softmax
    __shared__ __align__(16) float sPred[ROWS_PER_BLK][4];
    __shared__ __align__(16) float sTgt [COLS_PER_BLK][4];
    __shared__              int    sLab [COLS_PER_BLK];

    const int tid   = threadIdx.x;
    const int colG0 = blockIdx.x * COLS_PER_BLK;   // 5 column groups
    const int rowG0 = blockIdx.y * ROWS_PER_BLK;   // 500 row groups

    // ---- Stage target shapes + labels (coalesced b128 / b32) ----
    for (int t = tid; t < COLS_PER_BLK; t += 256) {
        *(v4f*)&sTgt[t][0] = *(const v4f*)(tgtS + (size_t)(colG0 + t) * 4);
        sLab[t] = tgtL[colG0 + t];
    }
    // ---- Stage predicted shapes for our 64 rows ----
    for (int r = tid; r < ROWS_PER_BLK; r += 256) {
        *(v4f*)&sPred[r][0] = *(const v4f*)(predS + (size_t)(rowG0 + r) * 4);
    }
    // ---- Negated softmax for our 64 rows (one thread per row; amortized over 64x640 outputs) ----
    if (tid < ROWS_PER_BLK) {
        const float* lg = logits + (size_t)(rowG0 + tid) * NCLS;
        float m = -3.402823466e38f;
        for (int k = 0; k < NCLS; ++k) m = fmaxf(m, lg[k]);
        float s = 0.0f;
        for (int k = 0; k < NCLS; ++k) s += __expf(lg[k] - m);
        const float ninv = -1.0f / s;
        for (int k = 0; k < NCLS; ++k) sProb[tid][k] = __expf(lg[k] - m) * ninv;
    }
    __syncthreads();

    const int wave = tid >> 5;        // 8 waves per block
    const int lane = tid & 31;
    const int nloc = lane & 15;       // N within 16x16 tile
    const int hi   = lane >> 4;       // 0: lanes 0-15, 1: lanes 16-31

    // Each wave owns 5 column tiles (ct = wave + 8j); per column tile it drives all
    // 4 row tiles as independent accumulator chains. Uniform trip counts across the
    // block => EXEC all-1s at every WMMA.
    for (int j = 0; j < 5; ++j) {
        const int ct   = wave + 8 * j;
        const int c0L  = ct * 16;
        const int colL = c0L + nloc;

        const float t0 = sTgt[colL][0], t1 = sTgt[colL][1];
        const float t2 = sTgt[colL][2], t3 = sTgt[colL][3];
        const int   lab  = sLab[colL];
        const int   labx = lab - 2 * hi;   // (labx == s4)   <=> lab == k0
        const int   laby = labx - 1;       // (laby == s4)   <=> lab == k0+1

        // ---- L1 cost straight into the 4 WMMA accumulators (C-matrix layout:
        //      VGPR vi, lane l -> M = vi + 8*hi, N = nloc) ----
        v8f acc[4];
#pragma unroll
        for (int rt = 0; rt < 4; ++rt) {
#pragma unroll
            for (int vi = 0; vi < 8; ++vi) {
                const int rL = rt * 16 + vi + hi * 8;    // broadcast LDS reads per half-wave
                const float d0 = fabsf(sPred[rL][0] - t0);
                const float d1 = fabsf(sPred[rL][1] - t1);
                const float d2 = fabsf(sPred[rL][2] - t2);
                const float d3 = fabsf(sPred[rL][3] - t3);
                acc[rt][vi] = (d0 + d1) + (d2 + d3);
            }
        }

        // ---- Class cost: acc += (-prob) x onehot(labels), rolled k-step loop.
        //      A 16x4 frag: lanes 0-15 row M=nloc K={k0,k0+1}, lanes 16-31 K={k0+2,k0+3}.
        //      B 4x16 frag (mirror): built ONCE per k-step, reused by 4 row-tile WMMAs.
        //      Per-rt LDS addresses differ by a constant 16*PROB_PITCH*4 B -> one
        //      address VGPR + immediate ds offsets. ----
#pragma clang loop unroll(disable)
        for (int s4 = 0; s4 < NCLS; s4 += 4) {
            v2f b;
            b.x = (labx == s4) ? 1.0f : 0.0f;
            b.y = (laby == s4) ? 1.0f : 0.0f;
#pragma unroll
            for (int rt = 0; rt < 4; ++rt) {
                const v2f a = *(const v2f*)&sProb[rt * 16 + nloc][s4 + 2 * hi];
                acc[rt] = __builtin_amdgcn_wmma_f32_16x16x4_f32(
                              /*neg_a=*/false, a, /*neg_b=*/false, b,
                              /*c_mod=*/(short)0, acc[rt],
                              /*reuse_a=*/false, /*reuse_b=*/false);
            }
        }

        // ---- Store: per (rt,vi), lanes 0-15 / 16-31 each write one contiguous 64B row segment ----
        const int colGl = colG0 + colL;
#pragma unroll
        for (int rt = 0; rt < 4; ++rt) {
#pragma unroll
            for (int vi = 0; vi < 8; ++vi) {
                const int rowGl = rowG0 + rt * 16 + vi + hi * 8;
                out[(size_t)rowGl * TCOLS + colGl] = acc[rt][vi];
            }
        }
    }
}

extern "C" void kernel_launch(void* const* d_in, const int* in_sizes, int n_in,
                              void* d_out, int out_size, void* d_ws, size_t ws_size,
                              hipStream_t stream) {
    const float* logits = (const float*)d_in[0];   // pred_logits [32,1000,92]
    const float* predS  = (const float*)d_in[1];   // pred_shapes [32,1000,4]
    const float* tgtS   = (const float*)d_in[2];   // tgt_shapes  [3200,4]
    const int*   tgtL   = (const int*)  d_in[3];   // tgt_labels  [3200]
    float*       out    = (float*)d_out;           // C [32,1000,3200]

    dim3 grid(TCOLS / COLS_PER_BLK, BSQ / ROWS_PER_BLK);   // (5, 500)
    hungarian_cost_kernel<<<grid, 256, 0, stream>>>(logits, predS, tgtS, tgtL, out);
}